// GatedGCN_LSPE_22789096473263
// MI455X (gfx1250) — compile-verified
//
#include <hip/hip_runtime.h>
#include <hip/hip_bf16.h>
#include <cstdint>
#include <cstddef>

typedef __attribute__((ext_vector_type(2))) float v2f;
typedef __attribute__((ext_vector_type(8))) float v8f;

// =====================================================================
// WMMA fp32 GEMM:  Y[M, NT*16] = act( X[M, K1+K2] @ W^T + bias )
//  - X row r is gathered via idx (if non-null); first K1 cols from X1,
//    next K2 cols from X2 (fused concat for [hs | p[send]]).
//  - W is [NT*16, K1+K2] row-major (PyTorch Linear layout), staged in LDS.
//  - One wave -> 16x(NT*16) tile via V_WMMA_F32_16X16X4_F32 (fp32 pipe).
//  - Block = 256 threads = 8 waves = 128 rows.
// ACT: 0 = none, 1 = relu
// =====================================================================
template<int K1, int K2, int NT, int ACT>
__global__ __launch_bounds__(256)
void wmma_gemm_k(const float* __restrict__ X1, const float* __restrict__ X2,
                 const int* __restrict__ idx, const float* __restrict__ W,
                 const float* __restrict__ bias, float* __restrict__ Y, int M)
{
    constexpr int K    = K1 + K2;
    constexpr int HOUT = NT * 16;
    __shared__ float sW[HOUT * K];
    for (int t = threadIdx.x; t < HOUT * K; t += 256) sW[t] = W[t];
    __syncthreads();

    const int wave = threadIdx.x >> 5;
    const int lane = threadIdx.x & 31;
    const int r0   = blockIdx.x * 128 + wave * 16;
    const int mrow = lane & 15;          // A-matrix row handled by this lane
    const int koff = (lane >> 4) << 1;   // lanes 16-31 hold K+2,K+3

    const int r   = r0 + mrow;
    const int src = (r < M) ? (idx ? idx[r] : r) : 0;
    const float* a1 = X1 + (size_t)src * K1 + koff;
    const float* a2 = (K2 > 0 && X2) ? (X2 + (size_t)src * K2 + koff) : X1;

    v8f acc[NT];
#pragma unroll
    for (int t = 0; t < NT; ++t)
#pragma unroll
        for (int v = 0; v < 8; ++v) acc[t][v] = 0.f;

#pragma unroll
    for (int kk = 0; kk < K; kk += 4) {
        const float* ap = (K2 != 0 && kk >= K1) ? (a2 + (kk - K1)) : (a1 + kk);
        v2f a; a.x = ap[0]; a.y = ap[1];                 // global_load_b64 (gathered)
#pragma unroll
        for (int t = 0; t < NT; ++t) {
            const int wrow = t * 16 + mrow;              // B col n = lane&15
            v2f b;
            b.x = sW[wrow * K + kk + koff];              // ds_load_b64
            b.y = sW[wrow * K + kk + koff + 1];
            acc[t] = __builtin_amdgcn_wmma_f32_16x16x4_f32(
                false, a, false, b, (short)0, acc[t], false, false);
        }
    }

    // C/D layout: lane L -> col = L&15 (+tile*16); VGPR v -> row v (+8 for hi lanes)
    const int rowAdd = (lane >> 4) << 3;
    const int ncol   = lane & 15;
#pragma unroll
    for (int t = 0; t < NT; ++t) {
        const int c  = t * 16 + ncol;
        const float bv = bias ? bias[c] : 0.f;
#pragma unroll
        for (int v = 0; v < 8; ++v) {
            const int row = r0 + v + rowAdd;
            if (row < M) {
                float val = acc[t][v] + bv;
                if (ACT == 1) val = fmaxf(val, 0.f);
                Y[(size_t)row * HOUT + c] = val;
            }
        }
    }
}

// ---------------------------------------------------------------------
// eta: one wave per edge.  eta = sigmoid(lh[s] + lh[r] + le[k]);
// store eta_new = eta/rowsum and rowsum; accumulate BN column stats of
// eta via LDS ds_add_f32 atomics, flushed once per block.
// ---------------------------------------------------------------------
__global__ __launch_bounds__(256)
void eta_k(const float* __restrict__ lh, const float* __restrict__ le,
           const int* __restrict__ send, const int* __restrict__ rec,
           float* __restrict__ eta_new, float* __restrict__ rowsum,
           float* __restrict__ stats, int E)
{
    __shared__ float s[128];
    if (threadIdx.x < 128) s[threadIdx.x] = 0.f;
    __syncthreads();

    const int wave = threadIdx.x >> 5;
    const int lane = threadIdx.x & 31;
    const int k    = blockIdx.x * 8 + wave;
    if (k < E) {
        const int si = send[k], ri = rec[k];
        const size_t sb = (size_t)si * 64, rb = (size_t)ri * 64, kb = (size_t)k * 64;
        float v0 = lh[sb + lane]      + lh[rb + lane]      + le[kb + lane];
        float v1 = lh[sb + 32 + lane] + lh[rb + 32 + lane] + le[kb + 32 + lane];
        float e0 = 1.f / (1.f + __expf(-v0));
        float e1 = 1.f / (1.f + __expf(-v1));
        float sum = e0 + e1;
#pragma unroll
        for (int m = 16; m >= 1; m >>= 1) sum += __shfl_xor(sum, m, 32);
        eta_new[kb + lane]      = e0 / sum;
        eta_new[kb + 32 + lane] = e1 / sum;
        if (lane == 0) rowsum[k] = sum;
        atomicAdd(&s[lane],          e0);
        atomicAdd(&s[32 + lane],     e1);
        atomicAdd(&s[64 + lane],     e0 * e0);
        atomicAdd(&s[96 + lane],     e1 * e1);
    }
    __syncthreads();
    if (threadIdx.x < 128) atomicAdd(&stats[threadIdx.x], s[threadIdx.x]);
}

// stats[0..63] = colsum -> mean ; stats[64..127] = colsumsq -> rsqrt(var+eps)
__global__ void bn_finalize_k(float* __restrict__ stats, float count)
{
    int c = threadIdx.x;
    if (c < 64) {
        float mean = stats[c] / count;
        float var  = stats[64 + c] / count - mean * mean;
        stats[c]      = mean;
        stats[64 + c] = rsqrtf(var + 1e-5f);
    }
}

__global__ __launch_bounds__(256)
void fill_zero_k(float* __restrict__ p, size_t n)
{
    size_t i = (size_t)blockIdx.x * 256 + threadIdx.x;
    if (i < n) p[i] = 0.f;
}

// e += relu(bn(eta)),  eta = eta_new * rowsum
__global__ __launch_bounds__(256)
void e_update_k(float* __restrict__ e, const float* __restrict__ eta_new,
                const float* __restrict__ rowsum, const float* __restrict__ stats,
                const float* __restrict__ g, const float* __restrict__ b, size_t total)
{
    size_t i = (size_t)blockIdx.x * 256 + threadIdx.x;
    if (i >= total) return;
    const int col = (int)(i & 63);
    const size_t k = i >> 6;
    float eta = eta_new[i] * rowsum[k];
    float v = (eta - stats[col]) * stats[64 + col] * g[col] + b[col];
    e[i] += fmaxf(v, 0.f);
}

// agg[rec[k]] += hp[k] * eta_new[k]
__global__ __launch_bounds__(256)
void agg_scatter_k(const float* __restrict__ hp, const float* __restrict__ eta_new,
                   const int* __restrict__ rec, float* __restrict__ agg, size_t total)
{
    size_t i = (size_t)blockIdx.x * 256 + threadIdx.x;
    if (i >= total) return;
    const int col = (int)(i & 63);
    const size_t k = i >> 6;
    atomicAdd(&agg[(size_t)rec[k] * 64 + col], hp[i] * eta_new[i]);
}

// t = hp_send + agg[k] (k<N, faithful num_segments=E bug) ; BN stats of t
__global__ __launch_bounds__(256)
void t_stats_k(float* __restrict__ t, const float* __restrict__ agg,
               float* __restrict__ stats, int N, size_t total)
{
    __shared__ float s[128];
    if (threadIdx.x < 128) s[threadIdx.x] = 0.f;
    __syncthreads();
    size_t i = (size_t)blockIdx.x * 256 + threadIdx.x;
    if (i < total) {
        const int col = (int)(i & 63);
        const size_t k = i >> 6;
        float v = t[i] + ((k < (size_t)N) ? agg[i] : 0.f);
        t[i] = v;
        atomicAdd(&s[col], v);
        atomicAdd(&s[64 + col], v * v);
    }
    __syncthreads();
    if (threadIdx.x < 128) atomicAdd(&stats[threadIdx.x], s[threadIdx.x]);
}

// h_next[send[k]] = h[send[k]] + relu(bn(t[k]))
__global__ __launch_bounds__(256)
void h_update_k(const float* __restrict__ h_cur, float* __restrict__ h_nxt,
                const float* __restrict__ t, const int* __restrict__ send,
                const float* __restrict__ stats, const float* __restrict__ g,
                const float* __restrict__ b, size_t total)
{
    size_t i = (size_t)blockIdx.x * 256 + threadIdx.x;
    if (i >= total) return;
    const int col = (int)(i & 63);
    const size_t k = i >> 6;
    const size_t d = (size_t)send[k] * 64 + col;
    float v = (t[i] - stats[col]) * stats[64 + col] * g[col] + b[col];
    h_nxt[d] = h_cur[d] + fmaxf(v, 0.f);
}

// aggp[rec[k]] += lp[rec[k]] * eta_new[k]
__global__ __launch_bounds__(256)
void aggp_scatter_k(const float* __restrict__ lp, const float* __restrict__ eta_new,
                    const int* __restrict__ rec, float* __restrict__ aggp, size_t total)
{
    size_t i = (size_t)blockIdx.x * 256 + threadIdx.x;
    if (i >= total) return;
    const int col = (int)(i & 63);
    const size_t k = i >> 6;
    const size_t rbase = (size_t)rec[k] * 64 + col;
    atomicAdd(&aggp[rbase], lp[rbase] * eta_new[i]);
}

// p_next[send[k]] = p[send[k]] + tanh(p1h[send[k]] + aggp[k])
__global__ __launch_bounds__(256)
void p_update_k(const float* __restrict__ p_cur, float* __restrict__ p_nxt,
                const float* __restrict__ p1h, const float* __restrict__ aggp,
                const int* __restrict__ send, int N, size_t total)
{
    size_t i = (size_t)blockIdx.x * 256 + threadIdx.x;
    if (i >= total) return;
    const int col = (int)(i & 63);
    const size_t k = i >> 6;
    const size_t d = (size_t)send[k] * 64 + col;
    float arg = p1h[d] + ((k < (size_t)N) ? aggp[i] : 0.f);
    p_nxt[d] = p_cur[d] + tanhf(arg);
}

// graph pooling: h_agg[batch[n]] += h[n], p_agg[batch[n]] += p[n]
__global__ __launch_bounds__(256)
void pool_k(const float* __restrict__ h, const float* __restrict__ p,
            const int* __restrict__ batch, float* __restrict__ hagg,
            float* __restrict__ pagg, size_t total)
{
    size_t i = (size_t)blockIdx.x * 256 + threadIdx.x;
    if (i >= total) return;
    const int col = (int)(i & 63);
    const size_t n = i >> 6;
    const size_t d = (size_t)batch[n] * 64 + col;
    atomicAdd(&hagg[d], h[i]);
    atomicAdd(&pagg[d], p[i]);
}

// out[g] = x2[g] . r3W + r3b   (G=256, K=32)
__global__ void r3_k(const float* __restrict__ x2, const float* __restrict__ W,
                     const float* __restrict__ b, float* __restrict__ out, int Gn)
{
    int g = blockIdx.x * blockDim.x + threadIdx.x;
    if (g < Gn) {
        float s = b[0];
#pragma unroll
        for (int j = 0; j < 32; ++j) s += x2[g * 32 + j] * W[j];
        out[g] = s;
    }
}

static inline unsigned cdiv(size_t a, size_t b) { return (unsigned)((a + b - 1) / b); }

extern "C" void kernel_launch(void* const* d_in, const int* in_sizes, int n_in,
                              void* d_out, int out_size, void* d_ws, size_t ws_size,
                              hipStream_t stream)
{
    const int N = in_sizes[0] / 128;      // 50000
    const int E = in_sizes[1] / 16;       // 800000
    const int Gn = 256, H = 64, L = 3;

    const float* h_in  = (const float*)d_in[0];
    const float* e_in  = (const float*)d_in[1];
    const float* p_in  = (const float*)d_in[2];
    const int*   eidx  = (const int*)d_in[3];
    const int*   batch = (const int*)d_in[4];
    const float* hembW = (const float*)d_in[5];  const float* hembB = (const float*)d_in[6];
    const float* eembW = (const float*)d_in[7];  const float* eembB = (const float*)d_in[8];
    const float* pembW = (const float*)d_in[9];  const float* pembB = (const float*)d_in[10];
    const float* linW  = (const float*)d_in[11]; const float* linB  = (const float*)d_in[12];
    const float* hpsW  = (const float*)d_in[13]; const float* hpsB  = (const float*)d_in[14];
    const float* hprW  = (const float*)d_in[15]; const float* hprB  = (const float*)d_in[16];
    const float* p1W   = (const float*)d_in[17]; const float* p1B   = (const float*)d_in[18];
    const float* p2W   = (const float*)d_in[19]; const float* p2B   = (const float*)d_in[20];
    const float* bnG   = (const float*)d_in[21]; const float* bnB   = (const float*)d_in[22];
    const float* r1W   = (const float*)d_in[23]; const float* r1B   = (const float*)d_in[24];
    const float* r2W   = (const float*)d_in[25]; const float* r2B   = (const float*)d_in[26];
    const float* r3W   = (const float*)d_in[27]; const float* r3B   = (const float*)d_in[28];

    const int* send = eidx;
    const int* rec  = eidx + E;

    // ---- workspace carve ----
    float* ws = (float*)d_ws; size_t off = 0;
    auto alloc = [&](size_t n) { float* p = ws + off; off += n; return p; };
    const size_t NH = (size_t)N * H, EH = (size_t)E * H;
    float* h0   = alloc(NH); float* h1 = alloc(NH);
    float* p0   = alloc(NH); float* p1 = alloc(NH);
    float* e_ws = alloc(EH);
    float* lh   = alloc(NH);
    float* buf1 = alloc(EH);          // le -> hp_rec
    float* buf2 = alloc(EH);          // hp_send -> t
    float* buf3 = alloc(EH);          // eta_new
    float* rsum = alloc((size_t)E);
    float* agg  = alloc(NH);
    float* aggp = alloc(NH);
    float* lp   = alloc(NH);
    float* p1h  = alloc(NH);
    float* stA  = alloc(128);         // eta BN stats
    float* stB  = alloc(128);         // t BN stats
    float* hagg = alloc((size_t)Gn * H);
    float* pagg = alloc((size_t)Gn * H);
    float* x1   = alloc((size_t)Gn * H);
    float* x2   = alloc((size_t)Gn * 32);
    (void)ws_size; (void)n_in; (void)out_size;

    const size_t EHtot = EH, NHtot = NH;

    // ---- embeddings (fp32 WMMA) ----
    wmma_gemm_k<128, 0, 4, 0><<<cdiv(N, 128), 256, 0, stream>>>(h_in, nullptr, nullptr, hembW, hembB, h0, N);
    wmma_gemm_k<16, 0, 4, 0><<<cdiv(E, 128), 256, 0, stream>>>(e_in, nullptr, nullptr, eembW, eembB, e_ws, E);
    wmma_gemm_k<16, 0, 4, 0><<<cdiv(N, 128), 256, 0, stream>>>(p_in, nullptr, nullptr, pembW, pembB, p0, N);

    float* h_cur = h0; float* h_nxt = h1;
    float* p_cur = p0; float* p_nxt = p1;

    for (int i = 0; i < L; ++i) {
        const float* lW  = linW + (size_t)i * H * H;   const float* lB  = linB + (size_t)i * H;
        const float* sW  = hpsW + (size_t)i * H * 2*H; const float* sB  = hpsB + (size_t)i * H;
        const float* rW  = hprW + (size_t)i * H * 2*H; const float* rB  = hprB + (size_t)i * H;
        const float* w1  = p1W  + (size_t)i * H * H;   const float* b1  = p1B  + (size_t)i * H;
        const float* w2  = p2W  + (size_t)i * H * H;   const float* b2  = p2B  + (size_t)i * H;
        const float* g   = bnG  + (size_t)i * H;       const float* bb  = bnB  + (size_t)i * H;

        // lh = lin(h) [N,64] ; le = lin(e) [E,64]  (bias once each -> 3b total in eta)
        wmma_gemm_k<64, 0, 4, 0><<<cdiv(N, 128), 256, 0, stream>>>(h_cur, nullptr, nullptr, lW, lB, lh, N);
        wmma_gemm_k<64, 0, 4, 0><<<cdiv(E, 128), 256, 0, stream>>>(e_ws, nullptr, nullptr, lW, lB, buf1, E);

        // eta_new + rowsum + eta BN stats
        fill_zero_k<<<1, 256, 0, stream>>>(stA, 128);
        eta_k<<<cdiv(E, 8), 256, 0, stream>>>(lh, buf1, send, rec, buf3, rsum, stA, E);
        bn_finalize_k<<<1, 64, 0, stream>>>(stA, (float)E);

        // e += relu(bn(eta))
        e_update_k<<<cdiv(EHtot, 256), 256, 0, stream>>>(e_ws, buf3, rsum, stA, g, bb, EHtot);

        // hp_rec = [h[rec] | p[rec]] @ hprW^T + b   (gather fused into WMMA A-loads)
        wmma_gemm_k<64, 64, 4, 0><<<cdiv(E, 128), 256, 0, stream>>>(h_cur, p_cur, rec, rW, rB, buf1, E);
        fill_zero_k<<<cdiv(NHtot, 256), 256, 0, stream>>>(agg, NHtot);
        agg_scatter_k<<<cdiv(EHtot, 256), 256, 0, stream>>>(buf1, buf3, rec, agg, EHtot);

        // hp_send, then t = hp_send + agg[k] with BN stats
        wmma_gemm_k<64, 64, 4, 0><<<cdiv(E, 128), 256, 0, stream>>>(h_cur, p_cur, send, sW, sB, buf2, E);
        fill_zero_k<<<1, 256, 0, stream>>>(stB, 128);
        t_stats_k<<<cdiv(EHtot, 256), 256, 0, stream>>>(buf2, agg, stB, N, EHtot);
        bn_finalize_k<<<1, 64, 0, stream>>>(stB, (float)E);

        // h scatter-set (double buffer = faithful .at[send].set semantics)
        hipMemcpyAsync(h_nxt, h_cur, NH * sizeof(float), hipMemcpyDeviceToDevice, stream);
        h_update_k<<<cdiv(EHtot, 256), 256, 0, stream>>>(h_cur, h_nxt, buf2, send, stB, g, bb, EHtot);

        // positional channel
        wmma_gemm_k<64, 0, 4, 0><<<cdiv(N, 128), 256, 0, stream>>>(p_cur, nullptr, nullptr, w2, b2, lp, N);
        fill_zero_k<<<cdiv(NHtot, 256), 256, 0, stream>>>(aggp, NHtot);
        aggp_scatter_k<<<cdiv(EHtot, 256), 256, 0, stream>>>(lp, buf3, rec, aggp, EHtot);
        wmma_gemm_k<64, 0, 4, 0><<<cdiv(N, 128), 256, 0, stream>>>(p_cur, nullptr, nullptr, w1, b1, p1h, N);
        hipMemcpyAsync(p_nxt, p_cur, NH * sizeof(float), hipMemcpyDeviceToDevice, stream);
        p_update_k<<<cdiv(EHtot, 256), 256, 0, stream>>>(p_cur, p_nxt, p1h, aggp, send, N, EHtot);

        { float* t = h_cur; h_cur = h_nxt; h_nxt = t; }
        { float* t = p_cur; p_cur = p_nxt; p_nxt = t; }
    }

    // ---- readout ----
    fill_zero_k<<<cdiv((size_t)Gn * H, 256), 256, 0, stream>>>(hagg, (size_t)Gn * H);
    fill_zero_k<<<cdiv((size_t)Gn * H, 256), 256, 0, stream>>>(pagg, (size_t)Gn * H);
    pool_k<<<cdiv(NHtot, 256), 256, 0, stream>>>(h_cur, p_cur, batch, hagg, pagg, NHtot);

    // r1: [256,128] @ [64,128]^T, relu  (concat fused); r2: [256,64] @ [32,64]^T, relu
    wmma_gemm_k<64, 64, 4, 1><<<cdiv(Gn, 128), 256, 0, stream>>>(hagg, pagg, nullptr, r1W, r1B, x1, Gn);
    wmma_gemm_k<64, 0, 2, 1><<<cdiv(Gn, 128), 256, 0, stream>>>(x1, nullptr, nullptr, r2W, r2B, x2, Gn);
    r3_k<<<1, 256, 0, stream>>>(x2, r3W, r3B, (float*)d_out, Gn);
}